// DigitCaps_8134668058920
// MI455X (gfx1250) — compile-verified
//
#include <hip/hip_runtime.h>

#define B_SZ 256
#define R_SZ 2592   // 32*9*9
#define L_SZ 32
#define C_SZ 16

typedef __attribute__((ext_vector_type(2))) float v2f;
typedef __attribute__((ext_vector_type(8))) float v8f;

// ---------------------------------------------------------------------------
// Kernel 1: u_hat[b,r,l] = sum_c W[r,l,c] * x[b,r,c]
// One block (8 waves) per r. Each wave owns 32 rows of b and all 32 columns l,
// i.e. 2 M-tiles x 2 N-tiles of 16x16, K=16 via 4 chained V_WMMA_F32_16X16X4_F32.
// W tile staged in LDS, padded to stride 18 (even -> 8B aligned pairs,
// 18*k mod 64 hits 16 distinct banks for 16 lanes -> conflict-free).
// ---------------------------------------------------------------------------
__global__ __launch_bounds__(256) void uhat_wmma_kernel(
    const float* __restrict__ x,   // (B, R, C)
    const float* __restrict__ W,   // (R, L, C)  (leading 1-dims squashed)
    float* __restrict__ u)         // (B, R, L)  -> u_j region of d_out
{
    __shared__ float sW[L_SZ * 18];
    const int r = blockIdx.x;
    const int t = threadIdx.x;

    for (int i = t; i < L_SZ * C_SZ; i += 256) {
        int l = i >> 4, c = i & 15;
        sW[l * 18 + c] = W[(size_t)r * (L_SZ * C_SZ) + i];
    }
    __syncthreads();

    const int wave = t >> 5;
    const int lane = t & 31;
    const int hi   = lane >> 4;   // 0: K pair {0,1}; 1: K pair {2,3}
    const int ln   = lane & 15;
    const int m0   = wave * 32;

    // A fragments (16x4 f32 layout): lane ln holds row m0+mt*16+ln,
    // vgpr pair = x[row][q*4 + 2*hi .. +1]  -> one 8B load each.
    v2f afrag[2][4];
    for (int mt = 0; mt < 2; ++mt) {
        const int row = m0 + mt * 16 + ln;
        const float* xr = x + (size_t)row * (R_SZ * C_SZ) + (size_t)r * C_SZ;
        for (int q = 0; q < 4; ++q) {
            const int kb = q * 4 + hi * 2;
            afrag[mt][q] = *(const v2f*)(xr + kb);
        }
    }

    // B fragments: B[k][n] = W[n][k] (transpose of W storage).
    v2f bfrag[2][4];
    for (int nt = 0; nt < 2; ++nt) {
        const int n = nt * 16 + ln;
        for (int q = 0; q < 4; ++q) {
            const int kb = q * 4 + hi * 2;
            v2f bb;
            bb.x = sW[n * 18 + kb];
            bb.y = sW[n * 18 + kb + 1];
            bfrag[nt][q] = bb;
        }
    }

    for (int mt = 0; mt < 2; ++mt) {
        for (int nt = 0; nt < 2; ++nt) {
            v8f acc = {};
            for (int q = 0; q < 4; ++q) {
                acc = __builtin_amdgcn_wmma_f32_16x16x4_f32(
                    /*neg_a=*/false, afrag[mt][q],
                    /*neg_b=*/false, bfrag[nt][q],
                    /*c_mod=*/(short)0, acc,
                    /*reuse_a=*/false, /*reuse_b=*/false);
            }
            // D layout: vgpr i, lanes 0-15 -> M=i, lanes 16-31 -> M=i+8; N=ln.
            const int row = m0 + mt * 16 + hi * 8;
            const int col = nt * 16 + ln;
            float* ur = u + (size_t)row * (R_SZ * L_SZ) + (size_t)r * L_SZ + col;
            for (int i = 0; i < 8; ++i)
                ur[(size_t)i * (R_SZ * L_SZ)] = acc[i];
        }
    }
}

// ---------------------------------------------------------------------------
__global__ __launch_bounds__(256) void init_b_kernel(float* __restrict__ blog)
{
    int i = blockIdx.x * 256 + threadIdx.x;
    if (i < R_SZ) blog[i] = 0.0f;
}

// c = softmax(b) over R, single block.
__global__ __launch_bounds__(256) void softmax_kernel(
    const float* __restrict__ blog, float* __restrict__ c)
{
    __shared__ float red[256];
    const int t = threadIdx.x;

    float m = -3.0e38f;
    for (int i = t; i < R_SZ; i += 256) m = fmaxf(m, blog[i]);
    red[t] = m; __syncthreads();
    for (int off = 128; off > 0; off >>= 1) {
        if (t < off) red[t] = fmaxf(red[t], red[t + off]);
        __syncthreads();
    }
    const float mx = red[0];
    __syncthreads();

    float s = 0.0f;
    for (int i = t; i < R_SZ; i += 256) s += expf(blog[i] - mx);
    red[t] = s; __syncthreads();
    for (int off = 128; off > 0; off >>= 1) {
        if (t < off) red[t] += red[t + off];
        __syncthreads();
    }
    const float inv = 1.0f / red[0];

    for (int i = t; i < R_SZ; i += 256) c[i] = expf(blog[i] - mx) * inv;
}

// s[b,l] = sum_r c[r]*u[b,r,l]; v = squash(s). One block per b.
// thread t: l = t&31, r-chunk = t>>5 -> fully coalesced 1KB reads per step.
// Each (b,r,l) element is read by exactly one thread, so on the FINAL
// iteration we fuse the output scaling u_j = c[r]*u_hat in place: the
// product c[r]*u is exactly the term accumulated into s, so the store is free
// (saves a whole separate 85MB read pass).
__global__ __launch_bounds__(256) void s_v_kernel(
    float* __restrict__ u, const float* __restrict__ c,
    float* __restrict__ v, float* __restrict__ vout, int scale_in_place)
{
    __shared__ float red[256];
    const int b = blockIdx.x;
    const int t = threadIdx.x;
    const int l = t & 31;
    const int chunk = t >> 5;

    float* ub = u + (size_t)b * (R_SZ * L_SZ);
    float acc = 0.0f;
    if (scale_in_place) {
        for (int r = chunk; r < R_SZ; r += 8) {
            const float term = c[r] * ub[(size_t)r * L_SZ + l];
            acc += term;
            ub[(size_t)r * L_SZ + l] = term;   // u_j = c[r]*u_hat
        }
    } else {
        for (int r = chunk; r < R_SZ; r += 8)
            acc = fmaf(c[r], ub[(size_t)r * L_SZ + l], acc);
    }

    red[t] = acc; __syncthreads();
    for (int off = 128; off >= 32; off >>= 1) {
        if (t < off) red[t] += red[t + off];
        __syncthreads();
    }
    if (t < 32) {
        const float s  = red[t];
        const float sq = s * s;
        const float vv = sq * s / ((1.0f + sq) * sqrtf(sq)); // matches reference squash
        v[b * L_SZ + t] = vv;
        if (vout) vout[b * L_SZ + t] = vv;
    }
}

// a[r] = mean_b sum_l u[b,r,l]*v[b,l]; b[r] += a[r]. One block per r.
__global__ __launch_bounds__(256) void a_update_kernel(
    const float* __restrict__ u, const float* __restrict__ v,
    float* __restrict__ blog)
{
    __shared__ float red[256];
    const int r = blockIdx.x;
    const int t = threadIdx.x;
    const int l = t & 31;
    const int chunk = t >> 5;

    float acc = 0.0f;
    for (int b = chunk; b < B_SZ; b += 8)
        acc = fmaf(u[(size_t)b * (R_SZ * L_SZ) + (size_t)r * L_SZ + l],
                   v[b * L_SZ + l], acc);

    red[t] = acc; __syncthreads();
    for (int off = 128; off > 0; off >>= 1) {
        if (t < off) red[t] += red[t + off];
        __syncthreads();
    }
    if (t == 0) blog[r] += red[0] * (1.0f / (float)B_SZ);
}

// ---------------------------------------------------------------------------
extern "C" void kernel_launch(void* const* d_in, const int* in_sizes, int n_in,
                              void* d_out, int out_size, void* d_ws, size_t ws_size,
                              hipStream_t stream)
{
    const float* x = (const float*)d_in[0];   // (B,R,C)
    const float* W = (const float*)d_in[1];   // (1,R,1,L,C)

    float* out  = (float*)d_out;
    float* vout = out;                                 // (B,1,L,1) = B*L floats
    float* u    = out + (size_t)B_SZ * L_SZ;           // (B,R,1,L,1) = B*R*L floats

    float* ws   = (float*)d_ws;
    float* blog = ws;              // R
    float* c    = ws + R_SZ;       // R
    float* v    = ws + 2 * R_SZ;   // B*L

    uhat_wmma_kernel<<<R_SZ, 256, 0, stream>>>(x, W, u);
    init_b_kernel<<<(R_SZ + 255) / 256, 256, 0, stream>>>(blog);

    for (int it = 0; it < 3; ++it) {
        const int last = (it == 2);
        softmax_kernel<<<1, 256, 0, stream>>>(blog, c);
        // On the last iteration, fuse u_j = c[r]*u_hat in place and emit v_out.
        s_v_kernel<<<B_SZ, 256, 0, stream>>>(u, c, v, last ? vout : nullptr, last);
        if (!last)
            a_update_kernel<<<R_SZ, 256, 0, stream>>>(u, v, blog);
    }
}